// SPDIncreaseDim_558345749053
// MI455X (gfx1250) — compile-verified
//
#include <hip/hip_runtime.h>
#include <stdint.h>

typedef float v2f __attribute__((ext_vector_type(2)));
typedef float v8f __attribute__((ext_vector_type(8)));
typedef unsigned int u32x4 __attribute__((ext_vector_type(4)));
typedef int          i32x4 __attribute__((ext_vector_type(4)));
typedef int          i32x8 __attribute__((ext_vector_type(8)));

#define BATCH 4096
#define N_IN  64
#define M_OUT 128

// LDS row strides in floats, padded so fragment reads dodge the 64 x 4B banks.
// 68 mod 64 = 4, 132 mod 64 = 4.
#define SX_LD 68
#define SE_LD 68
#define ST_LD 132

__global__ __launch_bounds__(256)
void spd_increase_dim_kernel(const float* __restrict__ X,
                             const float* __restrict__ E,    // [128 x 64]
                             const float* __restrict__ ADD,  // [128 x 128]
                             float* __restrict__ OUT)        // [B x 128 x 128]
{
    __shared__ float sX[N_IN  * SX_LD];   // X[b], 64x64  (TDM-loaded, HW-padded)
    __shared__ float sE[M_OUT * SE_LD];   // E, 128x64
    __shared__ float sT[N_IN  * ST_LD];   // tmp = X * E^T, 64x128

    const int tid  = threadIdx.x;
    const int wave = tid >> 5;
    const int lane = tid & 31;
    const int half = lane >> 4;   // 0: lanes 0-15, 1: lanes 16-31
    const int ln16 = lane & 15;
    const int kk   = half * 2;    // per-half K sub-offset inside a K=4 fragment

    const float* Xb = X + (size_t)blockIdx.x * N_IN * N_IN;

    // ---- wave 0: TDM DMA of X[b] (64x64 f32) into LDS with HW row padding ----
    if (wave == 0) {
        const uint64_t ga      = (uint64_t)(uintptr_t)Xb;
        const uint32_t lds_off = (uint32_t)(uintptr_t)sX;   // LDS byte offset

        // D# group 0: count=1 | lds_addr | global_addr[56:0] | type=2
        u32x4 g0;
        g0.x = 1u;                                   // count=1, user descriptor
        g0.y = lds_off;                              // LDS address (bytes)
        g0.z = (uint32_t)ga;                         // global_addr[31:0]
        g0.w = (uint32_t)(ga >> 32) | 0x80000000u;   // global_addr[56:32] | type=2

        // D# group 1: data_size=4B(2), pad_enable, pad_interval=64dw(5),
        // pad_amount=4dw(3); tensor 64x64, tile 64x64, row stride 64 elems.
        i32x8 g1;
        g1[0] = (2 << 16) | (1 << 20) | (5 << 22) | (3 << 25);
        g1[1] = (N_IN & 0xFFFF) << 16;               // tensor_dim0[15:0] = 64
        g1[2] = (N_IN & 0xFFFF) << 16;               // dim0 hi=0 | tensor_dim1[15:0]=64
        g1[3] = (N_IN & 0xFFFF) << 16;               // dim1 hi=0 | tile_dim0=64
        g1[4] = (N_IN & 0xFFFF);                     // tile_dim1=64, tile_dim2=0
        g1[5] = N_IN;                                // tensor_dim0_stride lo = 64
        g1[6] = (int)(((uint32_t)(N_IN * N_IN) & 0xFFFF) << 16); // dim1_stride lo16
        g1[7] = (int)((uint32_t)(N_IN * N_IN) >> 16);            // dim1_stride hi

        i32x4 gz4 = {0, 0, 0, 0};                    // groups 2/3 unused (2D)
        i32x8 gz8 = {0, 0, 0, 0, 0, 0, 0, 0};        // extra group (clang-23 form)
        __builtin_amdgcn_tensor_load_to_lds(g0, g1, gz4, gz4, gz8, 0);
        __builtin_amdgcn_s_wait_tensorcnt(0);
    } else {
        // ---- waves 1..7: stage E (128x64 f32) into padded LDS, overlapped ----
        const float4* src = (const float4*)E;        // 2048 float4
        const int t7 = tid - 32;                     // 0..223
        #pragma unroll
        for (int i = 0; i < 10; ++i) {               // 10*224 = 2240 >= 2048
            int v = t7 + i * 224;
            if (v < (M_OUT * N_IN) / 4) {
                int row = v >> 4;
                int c4  = (v & 15) << 2;
                float4 d = src[v];
                float* dst = &sE[row * SE_LD + c4];
                dst[0] = d.x; dst[1] = d.y; dst[2] = d.z; dst[3] = d.w;
            }
        }
    }
    __syncthreads();

    // ---- stage 1: tmp = X * E^T  (64 x 128) ----
    // 4x8 = 32 tiles of 16x16; wave w computes tiles [4w, 4w+4) (same mi,
    // so the X A-fragment is shared across its 4 tiles).
    #pragma unroll
    for (int t4 = 0; t4 < 4; ++t4) {
        const int t  = wave * 4 + t4;
        const int mi = t >> 3;   // row tile of X
        const int nj = t & 7;    // col tile of tmp
        v8f c = {0.f, 0.f, 0.f, 0.f, 0.f, 0.f, 0.f, 0.f};
        #pragma unroll
        for (int k = 0; k < N_IN; k += 4) {
            v2f a, bb;
            // A fragment: X rows mi*16..+15, K-slice [k, k+4)
            const float* ar = &sX[(mi * 16 + ln16) * SX_LD + k + kk];
            a.x = ar[0]; a.y = ar[1];
            // B fragment: (E^T)[k+kr][nj*16+n] = E[nj*16+n][k+kr]
            const float* br = &sE[(nj * 16 + ln16) * SE_LD + k + kk];
            bb.x = br[0]; bb.y = br[1];
            c = __builtin_amdgcn_wmma_f32_16x16x4_f32(
                    false, a, false, bb, (short)0, c, false, false);
        }
        #pragma unroll
        for (int r = 0; r < 8; ++r) {
            const int row = mi * 16 + r + 8 * half;  // C VGPR r <-> rows r, r+8
            sT[row * ST_LD + nj * 16 + ln16] = c[r];
        }
    }
    __syncthreads();

    // ---- stage 2: out[b] = E * tmp + add ----
    // wave w owns output rows [16w, 16w+16); hoist its E A-fragments (32 VGPRs)
    v2f aE[16];
    #pragma unroll
    for (int ks = 0; ks < 16; ++ks) {
        const float* ar = &sE[(wave * 16 + ln16) * SE_LD + ks * 4 + kk];
        aE[ks].x = ar[0]; aE[ks].y = ar[1];
    }

    float* outb = OUT + (size_t)blockIdx.x * M_OUT * M_OUT;
    for (int nj = 0; nj < 8; ++nj) {
        v8f c;
        #pragma unroll
        for (int r = 0; r < 8; ++r)
            c[r] = ADD[(wave * 16 + r + 8 * half) * M_OUT + nj * 16 + ln16];
        #pragma unroll
        for (int ks = 0; ks < 16; ++ks) {
            const int k = ks * 4;
            v2f bb;
            bb.x = sT[(k + kk    ) * ST_LD + nj * 16 + ln16];
            bb.y = sT[(k + kk + 1) * ST_LD + nj * 16 + ln16];
            c = __builtin_amdgcn_wmma_f32_16x16x4_f32(
                    false, aE[ks], false, bb, (short)0, c, false, false);
        }
        #pragma unroll
        for (int r = 0; r < 8; ++r)
            outb[(wave * 16 + r + 8 * half) * M_OUT + nj * 16 + ln16] = c[r];
    }
}

extern "C" void kernel_launch(void* const* d_in, const int* in_sizes, int n_in,
                              void* d_out, int out_size, void* d_ws, size_t ws_size,
                              hipStream_t stream) {
    const float* x   = (const float*)d_in[0];   // [4096, 64, 64]
    const float* eye = (const float*)d_in[1];   // [128, 64]
    const float* add = (const float*)d_in[2];   // [128, 128]
    float*       out = (float*)d_out;           // [4096, 128, 128]

    dim3 grid(BATCH);
    dim3 block(256);   // 8 wave32s: one workgroup per batch element
    hipLaunchKernelGGL(spd_increase_dim_kernel, grid, block, 0, stream,
                       x, eye, add, out);
}